// StaticCombiner_71141838291070
// MI455X (gfx1250) — compile-verified
//
#include <hip/hip_runtime.h>
#include <math.h>

// Problem constants (from reference)
#define VOCAB   32000
#define VOCAB4  (VOCAB / 4)     // 8000 float4 chunks per row
#define TOPK    16
#define NTHREADS 512            // 16 waves (wave32)
#define MIXW    0.3f
#define INV_BW  (1.0f / 100.0f)

typedef float v4f __attribute__((ext_vector_type(4)));

// One workgroup per row. Row (128 KB) is staged into LDS once via CDNA5
// async global->LDS copies; max / sum-exp / output passes all read LDS.
// HBM traffic per row: 128 KB read + 128 KB written => bandwidth-bound,
// ~45us floor at 23.3 TB/s for the whole grid.
__global__ __launch_bounds__(NTHREADS) void knn_lm_mix_kernel(
    const float* __restrict__ logits,        // (N, VOCAB)
    const float* __restrict__ distances,     // (N, TOPK)
    const int*   __restrict__ token_indices, // (N, TOPK)
    float* __restrict__ out)                 // (N, VOCAB)
{
    __shared__ float srow[VOCAB];   // 128000 B row stage (WGP LDS is 320 KB)
    __shared__ float red[32];       // cross-wave reduction scratch

    const int row  = blockIdx.x;
    const int tid  = threadIdx.x;
    const int lane = tid & 31;
    const int wid  = tid >> 5;

    const float* lrow = logits + (size_t)row * VOCAB;
    float*       orow = out    + (size_t)row * VOCAB;

    // ---- Stage: async global -> LDS copy of the whole row (1 HBM read) ----
    // Low 32 bits of a generic pointer to __shared__ == wave-relative LDS
    // byte address (shared aperture occupies the high bits only).
    unsigned lds_base = (unsigned)(size_t)(&srow[0]);
    unsigned long long gbase = (unsigned long long)lrow;
    for (int i = tid; i < VOCAB4; i += NTHREADS) {
        unsigned goff  = (unsigned)i * 16u;         // byte offset into row
        unsigned laddr = lds_base + goff;           // LDS byte address
        asm volatile("global_load_async_to_lds_b128 %0, %1, %2 offset:0"
                     :: "v"(laddr), "v"(goff), "s"(gbase)
                     : "memory");
    }
    asm volatile("s_wait_asynccnt 0" ::: "memory");
    __syncthreads();   // all waves waited their own ASYNCcnt before arriving

    const v4f* srow4 = (const v4f*)srow;

    // ---- Pass 1: row max (LDS read, no transcendentals) ----
    float m = -INFINITY;
    for (int i = tid; i < VOCAB4; i += NTHREADS) {
        v4f x = srow4[i];
        m = fmaxf(m, fmaxf(fmaxf(x.x, x.y), fmaxf(x.z, x.w)));
    }
#pragma unroll
    for (int off = 16; off > 0; off >>= 1)
        m = fmaxf(m, __shfl_xor(m, off, 32));
    if (lane == 0) red[wid] = m;
    __syncthreads();
    if (tid < 32) {
        float v = (tid < (NTHREADS / 32)) ? red[tid] : -INFINITY;
#pragma unroll
        for (int off = 16; off > 0; off >>= 1)
            v = fmaxf(v, __shfl_xor(v, off, 32));
        if (tid == 0) red[0] = v;
    }
    __syncthreads();
    const float M = red[0];
    __syncthreads();   // red[] about to be reused

    // ---- Pass 2: sum of exp(x - M) (1 transcendental per element) ----
    float s = 0.f;
    for (int i = tid; i < VOCAB4; i += NTHREADS) {
        v4f x = srow4[i];
        s += __expf(x.x - M) + __expf(x.y - M) +
             __expf(x.z - M) + __expf(x.w - M);
    }
#pragma unroll
    for (int off = 16; off > 0; off >>= 1)
        s += __shfl_xor(s, off, 32);
    if (lane == 0) red[wid] = s;
    __syncthreads();
    if (tid < 32) {
        float v = (tid < (NTHREADS / 32)) ? red[tid] : 0.f;
#pragma unroll
        for (int off = 16; off > 0; off >>= 1)
            v += __shfl_xor(v, off, 32);
        if (tid == 0) red[0] = v;
    }
    __syncthreads();
    const float S = red[0];

    // For positions without scatter contributions:
    //   log((1-MIXW) * exp(x - M) / S) = x + C
    const float C = __logf(1.0f - MIXW) - M - __logf(S);

    // ---- Pass 3: bulk output, one add per element, NT b128 stores ----
    for (int i = tid; i < VOCAB4; i += NTHREADS) {
        v4f x = srow4[i];
        v4f y;
        y.x = x.x + C; y.y = x.y + C; y.z = x.z + C; y.w = x.w + C;
        __builtin_nontemporal_store(y, (v4f*)orow + i);
    }

    // Make bulk stores visible before the sparse overwrite below.
    __threadfence();
    __syncthreads();

    // ---- Fix-up: the <=16 scattered vocab positions of this row ----
    if (tid < TOPK) {
        const float* drow = distances     + (size_t)row * TOPK;
        const int*   irow = token_indices + (size_t)row * TOPK;

        // Softmax over -d/100 across the 16 neighbors (computed redundantly
        // by each of the 16 lanes; all indices constant -> stays in VGPRs).
        float w[TOPK];
        int   idx[TOPK];
        float wm = -INFINITY;
#pragma unroll
        for (int j = 0; j < TOPK; j++) {
            float z = -drow[j] * INV_BW;
            w[j]   = z;
            idx[j] = irow[j];
            wm = fmaxf(wm, z);
        }
        float ws = 0.f;
#pragma unroll
        for (int j = 0; j < TOPK; j++) { w[j] = __expf(w[j] - wm); ws += w[j]; }

        // Accumulate duplicate indices; only the first occurrence writes.
        const int mine = irow[tid];
        float tot = 0.f;
        bool first = true;
#pragma unroll
        for (int j = 0; j < TOPK; j++) {
            if (idx[j] == mine) {
                tot += w[j];
                if (j < tid) first = false;
            }
        }
        if (first) {
            float x = srow[mine];                       // row still in LDS
            float p = __expf(x - M) * ((1.0f - MIXW) / S);
            orow[mine] = __logf(p + MIXW * (tot / ws));
        }
    }
}

extern "C" void kernel_launch(void* const* d_in, const int* in_sizes, int n_in,
                              void* d_out, int out_size, void* d_ws, size_t ws_size,
                              hipStream_t stream) {
    // setup_inputs order: hidden (unused), logits, distances, token_indices
    const float* logits = (const float*)d_in[1];
    const float* dist   = (const float*)d_in[2];
    const int*   tok    = (const int*)d_in[3];
    float*       out    = (float*)d_out;

    const int N = in_sizes[1] / VOCAB;   // rows = B*S
    knn_lm_mix_kernel<<<N, NTHREADS, 0, stream>>>(logits, dist, tok, out);
}